// PagedMinMaxPoolWrapper_59124519796899
// MI455X (gfx1250) — compile-verified
//
#include <hip/hip_runtime.h>
#include <cfloat>
#include <cstdint>

// Problem constants (match reference harness: B=4, MAX_BLOCKS=128, H=8, D=128)
#define TPB   64            // TOKENS_PER_BLOCK
#define SUBS  4             // SUB_CHUNK_PER_BLOCK
#define TPS   16            // TOKENS_PER_SUB_CHUNK
#define HD    (8 * 128)     // floats per token row (H*D)
#define ROW4  (HD / 4)      // float4 per token row = 256

static __device__ __forceinline__ float4 f4splat(float v) {
    return make_float4(v, v, v, v);
}

static __device__ __forceinline__ void async_b128_to_lds(uint32_t lds_byte_addr,
                                                         const void* gaddr) {
    // CDNA5 async copy engine: global -> LDS, tracked by ASYNCcnt.
    asm volatile("global_load_async_to_lds_b128 %0, %1, off"
                 :: "v"(lds_byte_addr), "v"(gaddr) : "memory");
}

static __device__ __forceinline__ void minmax4(float4& mn, float4& mx,
                                               const float4 v) {
    mn.x = fminf(mn.x, v.x); mx.x = fmaxf(mx.x, v.x);
    mn.y = fminf(mn.y, v.y); mx.y = fmaxf(mx.y, v.y);
    mn.z = fminf(mn.z, v.z); mx.z = fmaxf(mx.z, v.z);
    mn.w = fminf(mn.w, v.w); mx.w = fmaxf(mx.w, v.w);
}

__global__ __launch_bounds__(256) void paged_minmax_async_kernel(
    const float* __restrict__ keys,
    const int*   __restrict__ block_tables,
    const int*   __restrict__ cu_seqlens,
    const int*   __restrict__ heads_idx,
    float*       __restrict__ out,
    int B, int MB)
{
    // One 16-token tile of gathered rows: 16 tokens x 256 float4 = 64 KB LDS.
    __shared__ __align__(16) float4 smem[TPS * ROW4];

    const int nch = MB * SUBS;              // sub-chunks per batch (512)
    const int c   = blockIdx.x;             // sub-chunk (no division needed)
    const int b   = blockIdx.y;             // batch
    const int tid = threadIdx.x;            // 0..255: one float4 column of the row

    const int len = cu_seqlens[b + 1] - cu_seqlens[b];
    int valid = len - c * TPS;              // tokens of this sub-chunk inside seq
    valid = valid < 0 ? 0 : (valid > TPS ? TPS : valid);

    float4* out4 = reinterpret_cast<float4*>(out);
    const size_t obase = ((size_t)b * nch + c) * ROW4 + tid;
    float4* omin = out4 + obase;                                  // stack slot 0
    float4* omax = out4 + obase + (size_t)B * nch * ROW4;         // stack slot 1

    if (valid == 0) {                       // fully-masked chunk -> zeros
        float4 z = f4splat(0.0f);
        *omin = z;
        *omax = z;
        return;
    }

    // Paged indirection + head gather.
    const int blk   = block_tables[b * MB + (c >> 2)];
    const int tok0  = (c & (SUBS - 1)) * TPS;
    const int hcol  = tid >> 5;                  // output head (32 float4 per head)
    const int src_h = heads_idx[hcol];
    const int src_c = src_h * 32 + (tid & 31);   // float4 column in source row

    const float4* gbase =
        reinterpret_cast<const float4*>(keys) +
        ((size_t)blk * TPB + tok0) * ROW4 + src_c;

    // LDS byte address of this thread's slot for token 0 (generic addr low 32b).
    const uint32_t lbase = (uint32_t)(uintptr_t)(&smem[tid]);

    float4 mn = f4splat(FLT_MAX);
    float4 mx = f4splat(-FLT_MAX);

    if (valid == TPS) {
        // Fast path (vast majority of chunks): straight-line, branch-free.
        // 16 outstanding B128 async copies per wave, one counter wait, then
        // 16 ds_load_b128 reductions. EXEC all-1s throughout.
        #pragma unroll
        for (int t = 0; t < TPS; ++t) {
            async_b128_to_lds(lbase + (uint32_t)(t * (ROW4 * 16)),
                              (const void*)(gbase + (size_t)t * ROW4));
        }
        asm volatile("s_wait_asynccnt 0" ::: "memory");
        #pragma unroll
        for (int t = 0; t < TPS; ++t) {
            minmax4(mn, mx, smem[t * ROW4 + tid]);
        }
    } else {
        // Tail chunk of a sequence (at most one per batch): compact loop.
        // 'valid' is workgroup-uniform, so EXEC stays all-1s at every issue.
        for (int t = 0; t < valid; ++t) {
            async_b128_to_lds(lbase + (uint32_t)(t * (ROW4 * 16)),
                              (const void*)(gbase + (size_t)t * ROW4));
        }
        asm volatile("s_wait_asynccnt 0" ::: "memory");
        for (int t = 0; t < valid; ++t) {
            minmax4(mn, mx, smem[t * ROW4 + tid]);
        }
    }

    *omin = mn;
    *omax = mx;
}

extern "C" void kernel_launch(void* const* d_in, const int* in_sizes, int n_in,
                              void* d_out, int out_size, void* d_ws, size_t ws_size,
                              hipStream_t stream) {
    const float* keys         = (const float*)d_in[0];
    const int*   block_tables = (const int*)d_in[1];
    const int*   cu_seqlens   = (const int*)d_in[2];
    const int*   heads_idx    = (const int*)d_in[3];
    float*       out          = (float*)d_out;

    const int B   = in_sizes[2] - 1;        // 4
    const int MB  = in_sizes[1] / B;        // 128
    const int nch = MB * SUBS;              // 512 sub-chunks per batch

    dim3 grid(nch, B);                      // 2048 workgroups, no div in kernel
    dim3 block(256);                        // 8 wave32s: one float4 column each
    paged_minmax_async_kernel<<<grid, block, 0, stream>>>(
        keys, block_tables, cu_seqlens, heads_idx, out, B, MB);
}